// FeaturesLinear_3487513445027
// MI455X (gfx1250) — compile-verified
//
#include <hip/hip_runtime.h>
#include <hip/hip_bf16.h>

// FeaturesLinear: out[b] = bias + sum_f W[x[b,f] + offsets[f]]
// BATCH=65536, NUM_FIELDS=26, OUTPUT_DIM=1, W table = 4.16 MB (L2-resident).
// Gather/latency-bound: maximize outstanding global_load_b32 gathers; stage the
// sequential index stream into LDS with CDNA5 async global->LDS DMA (ASYNCcnt).

#define NF    26
#define BLOCK 256

#if defined(__gfx1250__) && \
    __has_builtin(__builtin_amdgcn_global_load_async_to_lds_b64) && \
    __has_builtin(__builtin_amdgcn_s_wait_asynccnt)
#define USE_ASYNC_LDS 1
#else
#define USE_ASYNC_LDS 0
#endif

#if USE_ASYNC_LDS
// b64 variant of the async builtin takes pointers to a 2-int vector.
typedef int v2i __attribute__((vector_size(8)));
typedef __attribute__((address_space(1))) v2i glob_v2i;   // global
typedef __attribute__((address_space(3))) v2i lds_v2i;    // LDS
#endif

__global__ __launch_bounds__(BLOCK) void FeaturesLinear_kernel(
    const int*   __restrict__ x,        // (batch, NF) int32
    const int*   __restrict__ offsets,  // (NF,) int32, uniform -> scalar loads
    const float* __restrict__ W,        // (FEATURE_DIM, 1) fp32
    const float* __restrict__ bias,     // (1,) fp32
    float*       __restrict__ out,      // (batch, 1) fp32
    int batch)
{
    const int tid  = threadIdx.x;
    const int row0 = blockIdx.x * BLOCK;
    const int row  = row0 + tid;

    int idx[NF];

#if USE_ASYNC_LDS
    // Stage this block's index tile (BLOCK rows * 104B = 26624B) into LDS with
    // fully coalesced async b64 DMA chunks. Each row is exactly 13 x 8B.
    __shared__ int s_idx[BLOCK * NF];
    const int rows  = (batch - row0 < BLOCK) ? (batch - row0) : BLOCK;
    const int* gsrc = x + (long long)row0 * NF;

    if (rows == BLOCK) {
        // Full tile (always, for batch % 256 == 0): no guards, one shared
        // vaddr (tid), 13 async instructions differing only in the immediate
        // offset — the HW applies IOFFSET to BOTH the global and LDS address.
        const glob_v2i* gp = (const glob_v2i*)(gsrc) + tid;
        lds_v2i*        lp = (lds_v2i*)(s_idx) + tid;
#define STAGE_CHUNK(K) \
        __builtin_amdgcn_global_load_async_to_lds_b64( \
            (glob_v2i*)gp, lp, (K) * BLOCK * 8, 0)
        STAGE_CHUNK(0);  STAGE_CHUNK(1);  STAGE_CHUNK(2);  STAGE_CHUNK(3);
        STAGE_CHUNK(4);  STAGE_CHUNK(5);  STAGE_CHUNK(6);  STAGE_CHUNK(7);
        STAGE_CHUNK(8);  STAGE_CHUNK(9);  STAGE_CHUNK(10); STAGE_CHUNK(11);
        STAGE_CHUNK(12);
#undef STAGE_CHUNK
    } else {
        // Tail tile (generic): guarded per-chunk.
        const int chunks = rows * (NF / 2);
        #pragma unroll
        for (int k = 0; k < (BLOCK * (NF / 2) + BLOCK - 1) / BLOCK; ++k) {
            int c = tid + k * BLOCK;
            if (c < chunks) {
                __builtin_amdgcn_global_load_async_to_lds_b64(
                    (glob_v2i*)(gsrc + c * 2),
                    (lds_v2i*)(&s_idx[c * 2]),
                    0, 0);
            }
        }
    }
    __builtin_amdgcn_s_wait_asynccnt(0);   // my wave's DMAs landed in LDS
    __syncthreads();                       // everyone's DMAs visible

    if (row < batch) {
        #pragma unroll
        for (int f = 0; f < NF; ++f) idx[f] = s_idx[tid * NF + f];
    }
#else
    if (row < batch) {
        const int2* r = (const int2*)(x + (long long)row * NF);
        #pragma unroll
        for (int i = 0; i < NF / 2; ++i) {
            int2 p = r[i];
            idx[2 * i]     = p.x;
            idx[2 * i + 1] = p.y;
        }
    }
#endif

    if (row >= batch) return;

    // Issue all 26 gathers before consuming any -> max loads in flight.
    float v[NF];
    #pragma unroll
    for (int f = 0; f < NF; ++f) {
        v[f] = W[idx[f] + offsets[f]];     // offsets[f]: uniform scalar load
    }

    // Two-accumulator reduction (26 fp32 adds; numerically benign at 0.01 scale).
    float s01 = 0.0f, s23 = 0.0f;
    #pragma unroll
    for (int f = 0; f < NF; f += 2) {
        s01 += v[f];
        s23 += v[f + 1];
    }
    out[row] = (s01 + s23) + bias[0];
}

extern "C" void kernel_launch(void* const* d_in, const int* in_sizes, int n_in,
                              void* d_out, int out_size, void* d_ws, size_t ws_size,
                              hipStream_t stream)
{
    const int*   x       = (const int*)  d_in[0];   // (B, 26) int32
    const int*   offsets = (const int*)  d_in[1];   // (26,)   int32
    const float* W       = (const float*)d_in[2];   // (1040000, 1) fp32
    const float* bias    = (const float*)d_in[3];   // (1,) fp32
    float*       out     = (float*)      d_out;     // (B, 1) fp32

    const int batch = in_sizes[0] / NF;             // 65536
    const int grid  = (batch + BLOCK - 1) / BLOCK;  // 256 blocks

    hipLaunchKernelGGL(FeaturesLinear_kernel, dim3(grid), dim3(BLOCK), 0, stream,
                       x, offsets, W, bias, out, batch);
}